// SparseLinear_11175504904588
// MI455X (gfx1250) — compile-verified
//
#include <hip/hip_runtime.h>

#define IN_DIM   4096
#define OUT_DIM  4096
#define NNZ_N    327680
#define BNNZ_N   2048
#define B_N      512
#define NTILES   (OUT_DIM / 16)   // 256 output-row tiles
#define KC       16               // k-slots per LDS chunk
#define SIN_STRIDE 520            // 512 + padding (bank-conflict avoidance, 16B aligned)

typedef float v2f __attribute__((ext_vector_type(2)));
typedef float v8f __attribute__((ext_vector_type(8)));
typedef int   v4i __attribute__((ext_vector_type(4)));

#if defined(__has_builtin)
#if __has_builtin(__builtin_amdgcn_global_load_async_to_lds_b128)
#define USE_ASYNC_LDS 1
#endif
#endif

__device__ __forceinline__ void async_copy16(const float* g, float* l) {
#if defined(USE_ASYNC_LDS)
    __builtin_amdgcn_global_load_async_to_lds_b128(
        (__attribute__((address_space(1))) v4i*)g,
        (__attribute__((address_space(3))) v4i*)l,
        0, 0);
#else
    *(float4*)l = *(const float4*)g;
#endif
}

__device__ __forceinline__ void wait_async_all() {
#if defined(USE_ASYNC_LDS)
#if __has_builtin(__builtin_amdgcn_s_wait_asynccnt)
    __builtin_amdgcn_s_wait_asynccnt(0);
#else
    asm volatile("s_wait_asynccnt 0" ::: "memory");
#endif
#endif
}

// ---------------- prep kernels ----------------

__global__ void k_zero(float* biasD, int* cnt) {
    int i = blockIdx.x * 256 + threadIdx.x;
    if (i < OUT_DIM) biasD[i] = 0.0f;
    if (i < NTILES)  cnt[i]   = 0;
}

// input [B, IN] (row-major) -> inT [IN, B] so column gathers are contiguous.
__global__ void k_transpose(const float* __restrict__ in, float* __restrict__ inT) {
    __shared__ float t[32][33];
    const int tx = threadIdx.x;           // 0..31
    const int ty = threadIdx.y;           // 0..7
    const int c0 = blockIdx.x * 32;       // along IN
    const int b0 = blockIdx.y * 32;       // along B
    #pragma unroll
    for (int j = 0; j < 32; j += 8)
        t[ty + j][tx] = in[(size_t)(b0 + ty + j) * IN_DIM + (c0 + tx)];
    __syncthreads();
    #pragma unroll
    for (int j = 0; j < 32; j += 8)
        inT[(size_t)(c0 + ty + j) * B_N + (b0 + tx)] = t[tx][ty + j];
}

__global__ void k_bias(const int* __restrict__ idx, const float* __restrict__ val,
                       float* biasD) {
    int i = blockIdx.x * 256 + threadIdx.x;
    if (i < BNNZ_N) atomicAdd(&biasD[idx[i]], val[i]);
}

__global__ void k_count(const int* __restrict__ rows, int* cnt) {
    int i = blockIdx.x * 256 + threadIdx.x;
    if (i < NNZ_N) atomicAdd(&cnt[rows[i] >> 4], 1);
}

// single block, NTILES threads: exclusive scan of tile counts
__global__ void k_scan(const int* __restrict__ cnt, int* off, int* cur) {
    __shared__ int s[NTILES];
    const int t = threadIdx.x;
    const int c = cnt[t];
    s[t] = c;
    __syncthreads();
    for (int d = 1; d < NTILES; d <<= 1) {
        int v = (t >= d) ? s[t - d] : 0;
        __syncthreads();
        s[t] += v;
        __syncthreads();
    }
    const int excl = s[t] - c;
    off[t] = excl;
    cur[t] = excl;
    if (t == NTILES - 1) off[NTILES] = s[t];
}

// bin nnz by 16-row tile: packed = (local_row << 16) | col, plus value array
__global__ void k_scatter(const int* __restrict__ rows, const int* __restrict__ cols,
                          const float* __restrict__ vals, int* cur,
                          int* packed, float* svals) {
    int i = blockIdx.x * 256 + threadIdx.x;
    if (i >= NNZ_N) return;
    const int r = rows[i];
    const int p = atomicAdd(&cur[r >> 4], 1);
    packed[p] = ((r & 15) << 16) | cols[i];
    svals[p]  = vals[i];
}

// ---------------- main WMMA kernel ----------------
// 1 workgroup (8 waves) per 16-row tile. Wave w owns batch rows [64w, 64w+64):
// four 16(b)x16(r) f32 accumulators driven by V_WMMA_F32_16X16X4_F32.
// Double-buffered LDS; gathers go through GLOBAL_LOAD_ASYNC_TO_LDS (ASYNCcnt)
// so chunk c+1's gather overlaps chunk c's WMMAs. One barrier per chunk.
__global__ __launch_bounds__(256) void k_spmm_wmma(
    const float* __restrict__ inT, const int* __restrict__ off,
    const int* __restrict__ packed, const float* __restrict__ svals,
    const float* __restrict__ biasD, float* __restrict__ out) {

    __shared__ float sIn[2][KC][SIN_STRIDE];  // gathered input columns [buf][k][b]
    __shared__ float sW[2][KC][16];           // one-hot ELL slice [buf][k][local_row]

    const int tid   = threadIdx.x;
    const int wave  = tid >> 5;
    const int lane  = tid & 31;
    const int l16   = lane & 15;
    const int khalf = (lane >> 4) << 1;    // 0 or 2 (A/B fragment K split)
    const int slot  = tid >> 4;            // 0..15 : k-slot this thread stages
    const int sub   = tid & 15;            // 0..15 : 16B piece / local row
    const int tile  = blockIdx.x;
    const int base  = off[tile];
    const int kTot  = off[tile + 1] - base;
    const int nCh   = (kTot + KC - 1) / KC;

    v8f acc0 = {}, acc1 = {}, acc2 = {}, acc3 = {};

    // stage chunk `kk` into buffer `buf`: sW one-hot + 16 async column gathers
    auto issue = [&](int buf, int kk) {
        const int idx = kk + slot;
        int col = 0, lr = 0;
        float v = 0.0f;
        if (idx < kTot) {
            const int p = packed[base + idx];
            col = p & 0xFFFF;
            lr  = p >> 16;
            v   = svals[base + idx];
        }
        sW[buf][slot][sub] = (sub == lr) ? v : 0.0f;
        const float* g = inT + (size_t)col * B_N;
        float* l = &sIn[buf][slot][0];
        #pragma unroll
        for (int j = 0; j < 8; ++j) {
            const int e = (sub + j * 16) << 2;   // float offset of this 16B piece
            async_copy16(g + e, l + e);
        }
    };

    if (nCh > 0) issue(0, 0);

    for (int c = 0; c < nCh; ++c) {
        wait_async_all();                  // own wave's chunk-c gathers landed
        __syncthreads();                   // all waves' gathers + sW visible;
                                           // nobody still reading buf (c+1)&1
        if (c + 1 < nCh) issue((c + 1) & 1, (c + 1) * KC);

        const int buf = c & 1;
        const int bw  = wave << 6;         // this wave's batch-row base
        #pragma unroll
        for (int k0 = 0; k0 < KC; k0 += 4) {
            v2f bf;
            bf.x = sW[buf][k0 + khalf + 0][l16];
            bf.y = sW[buf][k0 + khalf + 1][l16];

            v2f a;
            a.x = sIn[buf][k0 + khalf + 0][bw + 0 + l16];
            a.y = sIn[buf][k0 + khalf + 1][bw + 0 + l16];
            acc0 = __builtin_amdgcn_wmma_f32_16x16x4_f32(false, a, false, bf,
                                                         (short)0, acc0, false, false);
            a.x = sIn[buf][k0 + khalf + 0][bw + 16 + l16];
            a.y = sIn[buf][k0 + khalf + 1][bw + 16 + l16];
            acc1 = __builtin_amdgcn_wmma_f32_16x16x4_f32(false, a, false, bf,
                                                         (short)0, acc1, false, false);
            a.x = sIn[buf][k0 + khalf + 0][bw + 32 + l16];
            a.y = sIn[buf][k0 + khalf + 1][bw + 32 + l16];
            acc2 = __builtin_amdgcn_wmma_f32_16x16x4_f32(false, a, false, bf,
                                                         (short)0, acc2, false, false);
            a.x = sIn[buf][k0 + khalf + 0][bw + 48 + l16];
            a.y = sIn[buf][k0 + khalf + 1][bw + 48 + l16];
            acc3 = __builtin_amdgcn_wmma_f32_16x16x4_f32(false, a, false, bf,
                                                         (short)0, acc3, false, false);
        }
    }

    // epilogue: C/D layout -> lanes 0-15: N=lane, M=vgpr; lanes 16-31: N=lane-16, M=vgpr+8
    const int r  = (tile << 4) + l16;
    const float bv = biasD[r];
    const int bb = (wave << 6) + ((lane >> 4) << 3);
    #pragma unroll
    for (int i = 0; i < 8; ++i) {
        out[(size_t)(bb + i +  0) * OUT_DIM + r] = acc0[i] + bv;
        out[(size_t)(bb + i + 16) * OUT_DIM + r] = acc1[i] + bv;
        out[(size_t)(bb + i + 32) * OUT_DIM + r] = acc2[i] + bv;
        out[(size_t)(bb + i + 48) * OUT_DIM + r] = acc3[i] + bv;
    }
}

// ---------------- launch ----------------

extern "C" void kernel_launch(void* const* d_in, const int* in_sizes, int n_in,
                              void* d_out, int out_size, void* d_ws, size_t ws_size,
                              hipStream_t stream) {
    const float* input = (const float*)d_in[0];
    const int*   wrows = (const int*)  d_in[1];
    const int*   wcols = (const int*)  d_in[2];
    const float* wvals = (const float*)d_in[3];
    const int*   bidx  = (const int*)  d_in[4];
    const float* bvals = (const float*)d_in[5];
    float* out = (float*)d_out;

    char* ws = (char*)d_ws;
    float* inT    = (float*)ws;                                      // 8 MB
    float* biasD  = (float*)(ws + (size_t)IN_DIM * B_N * sizeof(float));
    int*   cnt    = (int*)((char*)biasD + OUT_DIM * sizeof(float));
    int*   off    = cnt + NTILES;                                    // NTILES+1
    int*   cur    = off + NTILES + 1;
    int*   packed = cur + NTILES;                                    // NNZ ints
    float* svals  = (float*)(packed + NNZ_N);                        // NNZ floats

    k_zero<<<(OUT_DIM + 255) / 256, 256, 0, stream>>>(biasD, cnt);

    dim3 tb(32, 8), tg(IN_DIM / 32, B_N / 32);
    k_transpose<<<tg, tb, 0, stream>>>(input, inT);

    k_bias<<<(BNNZ_N + 255) / 256, 256, 0, stream>>>(bidx, bvals, biasD);
    k_count<<<(NNZ_N + 255) / 256, 256, 0, stream>>>(wrows, cnt);
    k_scan<<<1, NTILES, 0, stream>>>(cnt, off, cur);
    k_scatter<<<(NNZ_N + 255) / 256, 256, 0, stream>>>(wrows, wcols, wvals,
                                                       cur, packed, svals);
    k_spmm_wmma<<<NTILES, 256, 0, stream>>>(inT, off, packed, svals, biasD, out);
}